// Net_71631464563092
// MI455X (gfx1250) — compile-verified
//
#include <hip/hip_runtime.h>

// ---------------------------------------------------------------------------
// out[n,c] = sum over 16 (128x128) blocks of max-over-block of
//            conv3x3(conv3x3(x, w1), w2)[n,c,:,:]   (maxpool folded into max)
// N=4, C=64, H=W=512, fp32 via V_WMMA_F32_16X16X4_F32 implicit GEMM.
// Staging uses GLOBAL_LOAD_ASYNC_TO_LDS_B32 (ASYNCcnt) for the input slab.
// ---------------------------------------------------------------------------

#define Nn 4
#define Cc 64
#define Hh 512
#define Ww 512
#define KSTEPS 144                         // (64/4) ci-steps * 9 taps
#define FRAG_FLOATS (4 * KSTEPS * 32 * 2)  // per weight set (4 co-blocks)

#define COLS 66        // 64 px + 2 halo columns
#define BLK  416       // dwords per ci-pair block: 3*66*2=396, pad->416 (==32 mod 64)

typedef __attribute__((ext_vector_type(2))) float v2f;
typedef __attribute__((ext_vector_type(8))) float v8f;

// monotonic float<->uint mapping so atomicMax on uint == float max
__device__ __forceinline__ unsigned ordf(float f) {
    unsigned u = __float_as_uint(f);
    return (u & 0x80000000u) ? ~u : (u | 0x80000000u);
}
__device__ __forceinline__ float unordf(unsigned u) {
    return __uint_as_float((u & 0x80000000u) ? (u & 0x7FFFFFFFu) : ~u);
}
#define ORD_NEG_INF 0x007FFFFFu   // ordf(-inf)

// ---------------------------------------------------------------------------
// Kernel 0: OIHW weights -> per-lane WMMA A-fragment layout.
// A 16x4 fp32 (ISA 7.12.2): lanes 0-15 = M rows with VGPR0=K0,VGPR1=K1;
// lanes 16-31 = same M with VGPR0=K2,VGPR1=K3.  s = (ci0/4)*9 + kh*3 + kw.
// Stored [cb][s][lane][2] floats -> one coalesced b64 per lane per step.
// ---------------------------------------------------------------------------
__global__ void reorg_weights(const float* __restrict__ w1,
                              const float* __restrict__ w2,
                              float* __restrict__ A1,
                              float* __restrict__ A2) {
    int idx = blockIdx.x * blockDim.x + threadIdx.x;
    const int total = 4 * KSTEPS * 32;
    if (idx >= 2 * total) return;
    const float* w = (idx < total) ? w1 : w2;
    float*       A = (idx < total) ? A1 : A2;
    int i    = idx % total;
    int lane = i & 31;
    int s    = (i >> 5) % KSTEPS;
    int cb   = i / (32 * KSTEPS);
    int ci0  = (s / 9) * 4;
    int r9   = s % 9;
    int m    = lane & 15;
    int kb   = (lane >> 4) * 2;
    int co   = cb * 16 + m;
    A[i * 2 + 0] = w[(co * Cc + ci0 + kb    ) * 9 + r9];
    A[i * 2 + 1] = w[(co * Cc + ci0 + kb + 1) * 9 + r9];
}

__global__ void init_blockbuf(unsigned* __restrict__ bb) {
    int i = blockIdx.x * blockDim.x + threadIdx.x;
    if (i < Nn * Cc * 16) bb[i] = ORD_NEG_INF;
}

// ---------------------------------------------------------------------------
// conv3x3 implicit GEMM, 'same' padding.
// Grid: x = wseg(8) + 8*h(512), y = n(4). Block 256 = 8 waves.
// wave = cb (4 co-blocks) + 4*pxb (two 32-px halves of the 64-px segment).
// Each wave computes TWO 16co x 16px tiles (A fragment reused across both).
// LDS slab layout (ci-pair interleaved so B loads are single ds_load_b64):
//   xs[cipair*BLK + (row*COLS + col)*2 + (ci&1)]
// ---------------------------------------------------------------------------
template <bool REDUCE>
__global__ __launch_bounds__(256) void conv3x3_wmma(
    const float* __restrict__ src,        // [N,64,512,512]
    const float* __restrict__ Afrags,     // reorg'd weights
    float* __restrict__ dst,              // y out (REDUCE=false)
    unsigned* __restrict__ blockbuf) {    // [N,64,16]     (REDUCE=true)

    __shared__ float xs[32 * BLK];        // 53,248 B

    const int n     = blockIdx.y;
    const int h     = blockIdx.x >> 3;
    const int wseg  = blockIdx.x & 7;
    const int wbase = wseg * 64;

    const int tid  = threadIdx.x;
    const int lane = tid & 31;
    const int wave = tid >> 5;
    const int cb   = wave & 3;            // co block
    const int pxb  = wave >> 2;           // 32-px half (0/1)
    const int co0  = cb * 16;

    // ---- async-stage halo'd slab: 64ci x rows[h-1..h+1] x cols[wbase-1..wbase+64]
    const float* sn = src + (size_t)n * Cc * Hh * Ww;
    for (int i = tid; i < Cc * 3 * COLS; i += 256) {
        int ci  = i / (3 * COLS);
        int rem = i - ci * (3 * COLS);
        int row = rem / COLS;
        int col = rem - row * COLS;
        int gh  = h - 1 + row;
        int gw  = wbase - 1 + col;
        int li  = (ci >> 1) * BLK + (row * COLS + col) * 2 + (ci & 1);
        if ((unsigned)gh < (unsigned)Hh && (unsigned)gw < (unsigned)Ww) {
            const float* gp = sn + ((size_t)ci * Hh + gh) * Ww + gw;
            // low 32 bits of a flat pointer into LDS == LDS byte offset
            unsigned lds_off = (unsigned)(uintptr_t)(&xs[li]);
            unsigned long long ga = (unsigned long long)(uintptr_t)gp;
            asm volatile("global_load_async_to_lds_b32 %0, %1, off"
                         :: "v"(lds_off), "v"(ga) : "memory");
        } else {
            xs[li] = 0.0f;                // zero padding halo
        }
    }
    asm volatile("s_wait_asynccnt 0x0" ::: "memory");
    __syncthreads();

    // ---- K loop: 16 ci-steps x 9 taps; 2 WMMA per A fragment
    v8f acc0 = {}, acc1 = {};
    const v2f* ap = (const v2f*)Afrags + (size_t)cb * KSTEPS * 32 + lane;
    const int colb = pxb * 32 + (lane & 15);  // sub-tile 0 column (before kw)
    const int cphi = lane >> 4;               // upper half-wave -> +1 ci pair

    for (int ci0 = 0; ci0 < Cc; ci0 += 4) {
        const float* xb = xs + ((ci0 >> 1) + cphi) * BLK;
        #pragma unroll
        for (int kh = 0; kh < 3; ++kh) {
            #pragma unroll
            for (int kw = 0; kw < 3; ++kw) {
                v2f a = *ap; ap += 32;
                int c0 = (kh * COLS + colb + kw) * 2;
                v2f b0 = *(const v2f*)(xb + c0);       // ds_load_b64
                v2f b1 = *(const v2f*)(xb + c0 + 32);  // +16 cols
                acc0 = __builtin_amdgcn_wmma_f32_16x16x4_f32(
                    false, a, false, b0, (short)0, acc0, false, false);
                acc1 = __builtin_amdgcn_wmma_f32_16x16x4_f32(
                    false, a, false, b1, (short)0, acc1, false, false);
            }
        }
    }

    // ---- epilogue. C/D layout: lanes 0-15: M=r; lanes 16-31: M=r+8.
    const int px   = lane & 15;
    const int half = lane >> 4;
    if (!REDUCE) {
        float* dp = dst + (((size_t)n * Cc + co0 + half * 8) * Hh + h) * Ww
                  + wbase + pxb * 32 + px;
        #pragma unroll
        for (int r = 0; r < 8; ++r) {
            dp[(size_t)r * Hh * Ww]      = acc0[r];
            dp[(size_t)r * Hh * Ww + 16] = acc1[r];
        }
    } else {
        // both sub-tiles lie in the same 128x128 block (wbase is 64-aligned)
        const int tile = ((h >> 7) << 2) + (wseg >> 1);
        #pragma unroll
        for (int r = 0; r < 8; ++r) {
            float v = fmaxf(acc0[r], acc1[r]);
            #pragma unroll
            for (int off = 1; off < 16; off <<= 1)
                v = fmaxf(v, __shfl_xor(v, off, 32));
            if ((lane & 15) == 0) {
                int ch = co0 + half * 8 + r;
                atomicMax(&blockbuf[(n * Cc + ch) * 16 + tile], ordf(v));
            }
        }
    }
}

__global__ void finalize_sum(const unsigned* __restrict__ bb,
                             float* __restrict__ out) {
    int i = threadIdx.x;                  // 0..255 == n*64+c
    float s = 0.0f;
    #pragma unroll
    for (int t = 0; t < 16; ++t) s += unordf(bb[i * 16 + t]);
    out[i] = s;
}

// ---------------------------------------------------------------------------
extern "C" void kernel_launch(void* const* d_in, const int* in_sizes, int n_in,
                              void* d_out, int out_size, void* d_ws, size_t ws_size,
                              hipStream_t stream) {
    (void)in_sizes; (void)n_in; (void)out_size; (void)ws_size;
    const float* x  = (const float*)d_in[0];
    const float* w1 = (const float*)d_in[1];
    const float* w2 = (const float*)d_in[2];
    float* out = (float*)d_out;

    char* ws = (char*)d_ws;
    const size_t y1_bytes = (size_t)Nn * Cc * Hh * Ww * sizeof(float);  // 268 MB
    float*    y1 = (float*)ws;
    float*    A1 = (float*)(ws + y1_bytes);
    float*    A2 = A1 + FRAG_FLOATS;
    unsigned* bb = (unsigned*)(A2 + FRAG_FLOATS);

    {
        int total = 2 * 4 * KSTEPS * 32;
        reorg_weights<<<(total + 255) / 256, 256, 0, stream>>>(w1, w2, A1, A2);
    }
    init_blockbuf<<<(Nn * Cc * 16 + 255) / 256, 256, 0, stream>>>(bb);

    dim3 grid(8 * Hh, Nn);
    conv3x3_wmma<false><<<grid, 256, 0, stream>>>(x,  A1, y1, nullptr);
    conv3x3_wmma<true ><<<grid, 256, 0, stream>>>(y1, A2, nullptr, bb);

    finalize_sum<<<1, Nn * Cc, 0, stream>>>(bb, out);
}